// Head_3539053052498
// MI455X (gfx1250) — compile-verified
//
#include <hip/hip_runtime.h>
#include <hip/hip_bf16.h>

#define Bb 4
#define Nn 4096
#define Hh 64
#define QT 256   // q-tile / threads per block in pass kernel
#define TK 128   // inner k tile (128-bit register mask)

typedef float v2f __attribute__((ext_vector_type(2)));
typedef float v8f __attribute__((ext_vector_type(8)));

// s = (Wq . Wk) / sqrt(H),  H = 64
__global__ void k_scale(const float* __restrict__ Wq, const float* __restrict__ Wk,
                        float* __restrict__ s) {
    if (threadIdx.x == 0) {
        float acc = 0.f;
#pragma unroll
        for (int h = 0; h < Hh; ++h) acc += Wq[h] * Wk[h];
        s[0] = acc * 0.125f;
    }
}

// One thread per q, all 4 batches; one block handles [q-tile] x [k-chunk].
// Emits per-(chunk,b,q) online-softmax partials (m, l, acc).
__global__ void __launch_bounds__(QT) k_pass(const float* __restrict__ X,
                                             const int* __restrict__ dag,
                                             const float* __restrict__ sptr,
                                             float* __restrict__ pm,
                                             float* __restrict__ pl,
                                             float* __restrict__ pa,
                                             int kchunk) {
    extern __shared__ float xs[];  // [Bb][kchunk]
    const int tid = threadIdx.x;
    const int q   = blockIdx.x * QT + tid;
    const int ch  = blockIdx.y;
    const int k0  = ch * kchunk;
    const float s = sptr[0];

    // stage this chunk of X (all batches) into LDS; later reads are broadcasts
    for (int i = tid; i < Bb * kchunk; i += QT) {
        int b = i / kchunk, k = i - b * kchunk;
        xs[i] = X[b * Nn + k0 + k];
    }
    __syncthreads();

    float c[Bb], M[Bb], L[Bb], A[Bb];
#pragma unroll
    for (int b = 0; b < Bb; ++b) {
        c[b] = s * X[b * Nn + q];     // score = c[b] * x[b][k]
        M[b] = 0.f; L[b] = 0.f; A[b] = 0.f;   // empty == (m=0,l=0,a=0): all finite
    }

    for (int t0 = 0; t0 < kchunk; t0 += TK) {
        const int* dp = dag + (size_t)(k0 + t0) * Nn + q;  // coalesced row segments
        unsigned msk[TK / 32];
        float mt[Bb];
#pragma unroll
        for (int b = 0; b < Bb; ++b) mt[b] = -__builtin_inff();

        // ---- phase A: load dag bits once, track per-batch max valid score ----
#pragma unroll
        for (int w = 0; w < TK / 32; ++w) {
            unsigned mw = 0;
#pragma unroll 4
            for (int kk = 0; kk < 32; ++kk) {
                int row = w * 32 + kk;
                if ((k0 + t0 + row + 16) < Nn)
                    __builtin_prefetch(dp + (size_t)(row + 16) * Nn, 0, 0);
                bool v = dp[(size_t)row * Nn] != 0;
                mw |= v ? (1u << kk) : 0u;
#pragma unroll
                for (int b = 0; b < Bb; ++b) {
                    float sc = c[b] * xs[b * kchunk + t0 + row];
                    mt[b] = v ? fmaxf(mt[b], sc) : mt[b];
                }
            }
            msk[w] = mw;
        }
        bool any = (msk[0] | msk[1] | msk[2] | msk[3]) != 0;
#pragma unroll
        for (int b = 0; b < Bb; ++b) mt[b] = any ? mt[b] : 0.f;  // keep finite

        // ---- phase B: one exp per (k,b); select (not mul) kills invalid lanes ----
        float lt[Bb] = {0, 0, 0, 0}, at[Bb] = {0, 0, 0, 0};
#pragma unroll
        for (int w = 0; w < TK / 32; ++w) {
            unsigned mw = msk[w];
#pragma unroll 4
            for (int kk = 0; kk < 32; ++kk) {
                bool v = (mw >> kk) & 1u;
#pragma unroll
                for (int b = 0; b < Bb; ++b) {
                    float xv = xs[b * kchunk + t0 + w * 32 + kk];
                    float e  = __expf(fmaf(c[b], xv, -mt[b]));  // <=1 when valid
                    float p  = v ? e : 0.f;                     // inf (invalid) discarded
                    lt[b] += p;
                    at[b]  = fmaf(p, xv, at[b]);
                }
            }
        }
        // ---- merge tile into chunk accumulators (all-finite LSE merge) ----
#pragma unroll
        for (int b = 0; b < Bb; ++b) {
            float nm = fmaxf(M[b], mt[b]);
            float r0 = __expf(M[b] - nm);
            float r1 = __expf(mt[b] - nm);
            L[b] = L[b] * r0 + lt[b] * r1;
            A[b] = A[b] * r0 + at[b] * r1;
            M[b] = nm;
        }
    }
#pragma unroll
    for (int b = 0; b < Bb; ++b) {
        size_t idx = ((size_t)ch * Bb + b) * Nn + q;   // coalesced over q
        pm[idx] = M[b]; pl[idx] = L[b]; pa[idx] = A[b];
    }
}

// One wave per (batch, 16 q's): merge chunk partials, then rank-1 epilogue
// out[16q x 64h] = y (16x1) * Wv (1x64) via four v_wmma_f32_16x16x4_f32.
__global__ void __launch_bounds__(32) k_reduce(const float* __restrict__ Wv,
                                               const float* __restrict__ pm,
                                               const float* __restrict__ pl,
                                               const float* __restrict__ pa,
                                               float* __restrict__ out,
                                               int CH) {
    const int lane = threadIdx.x;
    const int q0   = blockIdx.x * 16;
    const int b    = blockIdx.y;
    const int ql   = q0 + (lane & 15);   // lanes 16-31 duplicate 0-15: no divergence

    float M = 0.f, L = 0.f, A = 0.f;
    for (int ch = 0; ch < CH; ++ch) {
        size_t idx = ((size_t)ch * Bb + b) * Nn + ql;
        float mi = pm[idx], li = pl[idx], ai = pa[idx];
        float nm = fmaxf(M, mi);
        float r0 = __expf(M - nm);
        float r1 = __expf(mi - nm);
        L = L * r0 + li * r1;
        A = A * r0 + ai * r1;
        M = nm;
    }
    float y = (L > 0.f) ? (A / L) : 0.f;   // fully-masked rows -> 0 (NaN-safe path)

    // A-matrix 16x4: K=0 column = y, K=1..3 = 0  (VGPR0 lanes0-15 = K0, lanes16-31 = K2)
    v2f a;
    a.x = (lane < 16) ? y : 0.f;
    a.y = 0.f;
#pragma unroll
    for (int h0 = 0; h0 < Hh; h0 += 16) {
        // B-matrix 4x16: row K=0 = Wv[h0..h0+15], rows K=1..3 = 0
        v2f bv;
        bv.x = (lane < 16) ? Wv[h0 + lane] : 0.f;
        bv.y = 0.f;
        v8f cacc = {};
        cacc = __builtin_amdgcn_wmma_f32_16x16x4_f32(
            /*neg_a=*/false, a, /*neg_b=*/false, bv,
            /*c_mod=*/(short)0, cacc, /*reuse_a=*/false, /*reuse_b=*/false);
        // D layout: VGPR r -> M = r + 8*(lane>=16), N = lane&15
        int hh  = h0 + (lane & 15);
        int mof = (lane >> 4) * 8;
#pragma unroll
        for (int r = 0; r < 8; ++r) {
            out[((size_t)b * Nn + (q0 + mof + r)) * Hh + hh] = cacc[r];
        }
    }
}

extern "C" void kernel_launch(void* const* d_in, const int* in_sizes, int n_in,
                              void* d_out, int out_size, void* d_ws, size_t ws_size,
                              hipStream_t stream) {
    const float* X   = (const float*)d_in[0];
    const int*   dag = (const int*)d_in[1];
    const float* Wk  = (const float*)d_in[2];
    const float* Wq  = (const float*)d_in[3];
    const float* Wv  = (const float*)d_in[4];
    float* out = (float*)d_out;

    // pick #k-chunks that fits workspace: 3 * CH * B * N floats + header
    int CH = 32;
    while (CH > 2 && (256 + 3ull * CH * Bb * Nn * sizeof(float)) > ws_size) CH >>= 1;
    const int kchunk = Nn / CH;

    float* sptr = (float*)d_ws;
    float* pm   = (float*)((char*)d_ws + 256);
    float* pl   = pm + (size_t)CH * Bb * Nn;
    float* pa   = pl + (size_t)CH * Bb * Nn;

    k_scale<<<1, 64, 0, stream>>>(Wq, Wk, sptr);
    size_t lds = (size_t)Bb * kchunk * sizeof(float);
    k_pass<<<dim3(Nn / QT, CH), QT, lds, stream>>>(X, dag, sptr, pm, pl, pa, kchunk);
    k_reduce<<<dim3(Nn / 16, Bb), 32, 0, stream>>>(Wv, pm, pl, pa, out, CH);
}